// MapSampler2d_62251255988783
// MI455X (gfx1250) — compile-verified
//
#include <hip/hip_runtime.h>

typedef float v4f __attribute__((ext_vector_type(4)));
typedef int   v4i __attribute__((ext_vector_type(4)));

#define B_ 8
#define C_ 256
#define H_ 128
#define W_ 128
#define NPTS (1024 * 1024)   // G*P
#define WAVES 4
#define PTSW 8               // points per wave (software pipeline depth across these)

// ---------------- CDNA5 async global->LDS path (guarded) ----------------
#if defined(__AMDGCN__) && __has_builtin(__builtin_amdgcn_global_load_async_to_lds_b128) && __has_builtin(__builtin_amdgcn_s_wait_asynccnt)
#define HAVE_ASYNC 1
__device__ __forceinline__ void async_cp16(const float* g, float* l) {
    // signature: (v4i addrspace(1)* src, v4i addrspace(3)* dst, imm offset, imm cpol)
    __builtin_amdgcn_global_load_async_to_lds_b128(
        (__attribute__((address_space(1))) v4i*)g,
        (__attribute__((address_space(3))) v4i*)l,
        /*imm offset*/ 0, /*cpol*/ 0);
}
#else
#define HAVE_ASYNC 0
#endif

// ---------------- NCHW -> NHWC transpose into workspace ----------------
// in:  (B, C, H, W)   out: (B, H, W, C).  32x32 (c,x) tiles per (b,y) slice.
__global__ __launch_bounds__(256) void nchw_to_nhwc(const float* __restrict__ in,
                                                    float* __restrict__ out) {
    __shared__ float tile[32][33];
    const int tx = threadIdx.x;           // 0..31
    const int ty = threadIdx.y;           // 0..7
    const int xblk = blockIdx.x;          // W/32 = 4
    const int cblk = blockIdx.y;          // C/32 = 8
    const int by = blockIdx.z;            // B*H = 1024
    const int b = by >> 7;
    const int y = by & 127;
    const int x = xblk * 32 + tx;
#pragma unroll
    for (int i = 0; i < 4; ++i) {
        const int c = cblk * 32 + ty + i * 8;
        tile[ty + i * 8][tx] =
            __builtin_nontemporal_load(&in[(((long long)b * C_ + c) * H_ + y) * W_ + x]);
    }
    __syncthreads();
#pragma unroll
    for (int i = 0; i < 4; ++i) {
        const int xo = xblk * 32 + ty + i * 8;
        const int co = cblk * 32 + tx;
        out[((((long long)b * H_ + y) * W_ + xo)) * C_ + co] = tile[tx][ty + i * 8];
    }
}

// ---------------- bilinear weights helper ----------------
struct Corners {
    float w[4];
    int   off[4];   // (y*W + x) of each clamped corner
    int   bid;
};
__device__ __forceinline__ Corners make_corners(const float* __restrict__ pts,
                                                const int* __restrict__ bids,
                                                long long pt) {
    Corners r;
    const int g = (int)(pt >> 10);   // pt / P_ (P_ = 1024)
    const float u = pts[2 * pt + 0];
    const float v = pts[2 * pt + 1];
    const float x = u * (float)W_ - 0.5f;
    const float y = v * (float)H_ - 0.5f;
    const float xf = floorf(x), yf = floorf(y);
    const float wx = x - xf, wy = y - yf;
    const int x0 = (int)xf, y0 = (int)yf;
    const int x1 = x0 + 1, y1 = y0 + 1;
    const float vx0 = (x0 >= 0 && x0 < W_) ? 1.f : 0.f;
    const float vx1 = (x1 >= 0 && x1 < W_) ? 1.f : 0.f;
    const float vy0 = (y0 >= 0 && y0 < H_) ? 1.f : 0.f;
    const float vy1 = (y1 >= 0 && y1 < H_) ? 1.f : 0.f;
    const int xc0 = min(max(x0, 0), W_ - 1), xc1 = min(max(x1, 0), W_ - 1);
    const int yc0 = min(max(y0, 0), H_ - 1), yc1 = min(max(y1, 0), H_ - 1);
    r.w[0] = (1.f - wx) * (1.f - wy) * vx0 * vy0;
    r.w[1] = wx * (1.f - wy) * vx1 * vy0;
    r.w[2] = (1.f - wx) * wy * vx0 * vy1;
    r.w[3] = wx * wy * vx1 * vy1;
    r.off[0] = yc0 * W_ + xc0;
    r.off[1] = yc0 * W_ + xc1;
    r.off[2] = yc1 * W_ + xc0;
    r.off[3] = yc1 * W_ + xc1;
    r.bid = bids[g];
    return r;
}

// ---------------- main gather kernel (NHWC map in workspace) ----------------
// One wave handles one point at a time; lanes span channels (c = lane*4 and
// c = 128 + lane*4) -> fully coalesced 512B accesses. Double-buffered async
// staging of the 4 corner rows (4KB/point) into LDS pipelines global latency.
__global__ __launch_bounds__(WAVES * 32) void gather_nhwc(
    const float* __restrict__ map,   // (B,H,W,C)
    const float* __restrict__ pts,
    const int* __restrict__ bids,
    float* __restrict__ out) {
#if HAVE_ASYNC
    __shared__ float lds[WAVES][2][4 * C_];   // 32 KB
#endif
    const int lane = threadIdx.x & 31;
    const int wave = threadIdx.x >> 5;
    const long long pt0 = ((long long)blockIdx.x * WAVES + wave) * PTSW;

    float        wts[2][4];
    const float* cb[2][4];
    long long    ptb[2];

    auto stage = [&](long long pt, int buf) {
        const Corners cr = make_corners(pts, bids, pt);
        const float* base = map + (long long)cr.bid * (H_ * W_ * C_);
#pragma unroll
        for (int k = 0; k < 4; ++k) {
            wts[buf][k] = cr.w[k];
            cb[buf][k] = base + (long long)cr.off[k] * C_;
        }
        ptb[buf] = pt;
#if HAVE_ASYNC
#pragma unroll
        for (int k = 0; k < 4; ++k) {
            const float* src = cb[buf][k] + lane * 4;
            float* dst = &lds[wave][buf][k * C_ + lane * 4];
            async_cp16(src, dst);
            async_cp16(src + 128, dst + 128);
        }
#endif
    };

    auto compute = [&](int buf) {
        v4f acc0 = {0.f, 0.f, 0.f, 0.f};
        v4f acc1 = {0.f, 0.f, 0.f, 0.f};
#pragma unroll
        for (int k = 0; k < 4; ++k) {
            const float w = wts[buf][k];
#if HAVE_ASYNC
            v4f a = *(const v4f*)&lds[wave][buf][k * C_ + lane * 4];
            v4f b = *(const v4f*)&lds[wave][buf][k * C_ + 128 + lane * 4];
#else
            v4f a = *(const v4f*)(cb[buf][k] + lane * 4);
            v4f b = *(const v4f*)(cb[buf][k] + 128 + lane * 4);
#endif
            acc0 += a * w;
            acc1 += b * w;
        }
        float* o = out + ptb[buf] * C_;
        __builtin_nontemporal_store(acc0, (v4f*)(o + lane * 4));         // keep the NHWC
        __builtin_nontemporal_store(acc1, (v4f*)(o + 128 + lane * 4));   // map hot in L2
    };

    stage(pt0, 0);
#pragma unroll
    for (int j = 0; j < PTSW; ++j) {
        if (j + 1 < PTSW) {
            stage(pt0 + j + 1, (j + 1) & 1);
#if HAVE_ASYNC
            __builtin_amdgcn_s_wait_asynccnt(8);   // oldest batch of 8 complete
#endif
        } else {
#if HAVE_ASYNC
            __builtin_amdgcn_s_wait_asynccnt(0);
#endif
        }
        compute(j & 1);
    }
}

// ---------------- correctness fallback: direct NCHW gather ----------------
__global__ __launch_bounds__(256) void gather_nchw(const float* __restrict__ map,
                                                   const float* __restrict__ pts,
                                                   const int* __restrict__ bids,
                                                   float* __restrict__ out) {
    const int lane = threadIdx.x & 31;
    const long long pt = (long long)blockIdx.x * 8 + (threadIdx.x >> 5);
    const Corners cr = make_corners(pts, bids, pt);
    const long long bbase = (long long)cr.bid * C_ * H_ * W_;
#pragma unroll
    for (int i = 0; i < 8; ++i) {
        const int c = lane + 32 * i;
        const long long cbase = bbase + (long long)c * (H_ * W_);
        float v = map[cbase + cr.off[0]] * cr.w[0] + map[cbase + cr.off[1]] * cr.w[1] +
                  map[cbase + cr.off[2]] * cr.w[2] + map[cbase + cr.off[3]] * cr.w[3];
        __builtin_nontemporal_store(v, out + pt * C_ + c);
    }
}

extern "C" void kernel_launch(void* const* d_in, const int* in_sizes, int n_in,
                              void* d_out, int out_size, void* d_ws, size_t ws_size,
                              hipStream_t stream) {
    const float* map = (const float*)d_in[0];
    const float* pts = (const float*)d_in[1];
    const int*   bids = (const int*)d_in[2];
    float* out = (float*)d_out;

    const size_t need = (size_t)B_ * H_ * W_ * C_ * sizeof(float);   // 128 MB
    if (ws_size >= need) {
        float* nhwc = (float*)d_ws;
        dim3 tb(32, 8, 1);
        dim3 tg(W_ / 32, C_ / 32, B_ * H_);
        nchw_to_nhwc<<<tg, tb, 0, stream>>>(map, nhwc);
        const int blocks = NPTS / (WAVES * PTSW);
        gather_nhwc<<<blocks, WAVES * 32, 0, stream>>>(nhwc, pts, bids, out);
    } else {
        gather_nchw<<<NPTS / 8, 256, 0, stream>>>(map, pts, bids, out);
    }
}